// HaarPool_47493748359650
// MI455X (gfx1250) — compile-verified
//
#include <hip/hip_runtime.h>

#define NN 131072
#define GG 1024
#define HH 128
#define DEG 16
#define EE (NN * DEG)
#define LL 3
#define CC 10
#define BN_EPS 1e-5f
#define INV_SQRT2 0.7071067811865476f

typedef __attribute__((ext_vector_type(2))) float v2f;
typedef __attribute__((ext_vector_type(8))) float v8f;

// ---------------------------------------------------------------------------
// Y[nrows x 128] = X[nrows x 128] @ W[128 x 128] + bias   (fp32 WMMA 16x16x4)
// Block: 256 threads = 8 waves. Block owns 16 rows; wave w owns cols [16w,16w+16).
// A tile staged in LDS (padded stride 132 -> conflict-free reads).
// A layout per lane (ISA 7.12.2, 32-bit A 16x4): lanes 0-15 -> {K=0,K=1},
// lanes 16-31 -> {K=2,K=3}. B: row K striped across lanes within a VGPR.
// ---------------------------------------------------------------------------
__global__ void __launch_bounds__(256) gemm_bias_kernel(
    const float* __restrict__ X, const float* __restrict__ W,
    const float* __restrict__ bias, float* __restrict__ Y) {
  __shared__ float As[16][132];
  const int tid  = threadIdx.x;
  const int row0 = blockIdx.x << 4;
  for (int i = tid; i < 16 * 128; i += 256) {
    const int r = i >> 7, c = i & 127;
    As[r][c] = X[(size_t)(row0 + r) * HH + c];
  }
  __syncthreads();

  const int lane = tid & 31;
  const int half = lane >> 4;      // 0: K=0,1   1: K=2,3
  const int l    = lane & 15;
  const int col0 = (tid >> 5) << 4;

  v8f acc = {0.f, 0.f, 0.f, 0.f, 0.f, 0.f, 0.f, 0.f};
  #pragma unroll 4
  for (int kk = 0; kk < 32; ++kk) {
    const int kb = (kk << 2) + (half << 1);
    v2f a, b;
    a.x = As[l][kb];
    a.y = As[l][kb + 1];
    b.x = W[kb * HH + col0 + l];
    b.y = W[(kb + 1) * HH + col0 + l];
    acc = __builtin_amdgcn_wmma_f32_16x16x4_f32(
        /*neg_a=*/false, a, /*neg_b=*/false, b,
        /*c_mod=*/(short)0, acc, /*reuse_a=*/false, /*reuse_b=*/false);
  }
  #pragma unroll
  for (int r = 0; r < 8; ++r) {
    const int row = row0 + r + (half << 3);   // C/D: vgpr r -> M=r (lanes 0-15), M=r+8 (16-31)
    const int col = col0 + l;
    Y[(size_t)row * HH + col] = acc[r] + bias[col];
  }
}

__global__ void zero_kernel(float* __restrict__ p, int n) {
  const int i = blockIdx.x * blockDim.x + threadIdx.x;
  if (i < n) p[i] = 0.f;
}

// Per-column sum / sumsq partials: 128 threads (one per column), 64 rows/block.
__global__ void __launch_bounds__(128) bn_stats_kernel(
    const float* __restrict__ x, int ld, float* __restrict__ sums) {
  const int col = threadIdx.x;
  const int r0  = blockIdx.x * 64;
  float s = 0.f, s2 = 0.f;
  for (int r = 0; r < 64; ++r) {
    const float v = x[(size_t)(r0 + r) * ld + col];
    s += v;
    s2 = fmaf(v, v, s2);
  }
  atomicAdd(&sums[col], s);
  atomicAdd(&sums[HH + col], s2);
}

// x = relu((x - mean) * rsqrt(var + eps) * g + b), biased variance (jnp.var).
__global__ void bn_apply_relu_kernel(float* __restrict__ x, int ld,
                                     const float* __restrict__ sums,
                                     const float* __restrict__ gma,
                                     const float* __restrict__ bta, float inv_n) {
  const int i = blockIdx.x * blockDim.x + threadIdx.x;
  const int r = i >> 7, c = i & 127;
  const float m   = sums[c] * inv_n;
  const float var = fmaf(-m, m, sums[HH + c] * inv_n);
  const float v   = (x[(size_t)r * ld + c] - m) * rsqrtf(var + BN_EPS) * gma[c] + bta[c];
  x[(size_t)r * ld + c] = fmaxf(v, 0.f);
}

__global__ void copy4_kernel(float4* __restrict__ o, const float4* __restrict__ in, int n4) {
  const int i = blockIdx.x * blockDim.x + threadIdx.x;
  if (i < n4) o[i] = in[i];
}

// h[dst] += x[src] for every edge; one wave per edge, 4 floats per lane.
__global__ void __launch_bounds__(256) gin_scatter_kernel(
    const float* __restrict__ x, float* __restrict__ h,
    const int* __restrict__ src, const int* __restrict__ dst, int shift) {
  const int e    = (blockIdx.x << 3) + (threadIdx.x >> 5);
  const int lane = threadIdx.x & 31;
  const int s = src[e] >> shift;
  const int d = dst[e] >> shift;
  const float4 v = *(const float4*)(x + (size_t)s * HH + (lane << 2));
  float* hp = h + (size_t)d * HH + (lane << 2);
  atomicAdd(hp + 0, v.x);
  atomicAdd(hp + 1, v.y);
  atomicAdd(hp + 2, v.z);
  atomicAdd(hp + 3, v.w);
}

// y[k] = (x[2k] + x[2k+1]) / sqrt(2), rows of 128 floats (32 float4).
__global__ void haar_pool_kernel(const float4* __restrict__ x,
                                 float4* __restrict__ y, int n4out) {
  const int i = blockIdx.x * blockDim.x + threadIdx.x;
  if (i >= n4out) return;
  const int k = i >> 5, f = i & 31;
  const float4 a = x[(size_t)(k << 1) * 32 + f];
  const float4 b = x[((size_t)(k << 1) + 1) * 32 + f];
  float4 o;
  o.x = (a.x + b.x) * INV_SQRT2;
  o.y = (a.y + b.y) * INV_SQRT2;
  o.z = (a.z + b.z) * INV_SQRT2;
  o.w = (a.w + b.w) * INV_SQRT2;
  y[i] = o;
}

// embd[g, coloff+f] = sum over npg contiguous nodes of graph g.
__global__ void graph_pool_kernel(const float* __restrict__ x,
                                  float* __restrict__ embd, int npg, int coloff) {
  const int i = blockIdx.x * blockDim.x + threadIdx.x;   // GG*HH threads
  const int g = i >> 7, f = i & 127;
  const float* p = x + (size_t)g * npg * HH + f;
  float s = 0.f;
  for (int j = 0; j < npg; ++j) s += p[(size_t)j * HH];
  embd[(size_t)g * (LL * HH) + coloff + f] = s;
}

// out[g,c] = embd[g,:] . lin_w[:,c] + lin_b[c]   (384 -> 10)
__global__ void final_linear_kernel(const float* __restrict__ embd,
                                    const float* __restrict__ w,
                                    const float* __restrict__ b,
                                    float* __restrict__ out) {
  const int i = blockIdx.x * blockDim.x + threadIdx.x;
  if (i >= GG * CC) return;
  const int g = i / CC, c = i % CC;
  float s = b[c];
  for (int k = 0; k < LL * HH; ++k)
    s = fmaf(embd[(size_t)g * (LL * HH) + k], w[k * CC + c], s);
  out[i] = s;
}

// ---------------------------------------------------------------------------
extern "C" void kernel_launch(void* const* d_in, const int* in_sizes, int n_in,
                              void* d_out, int out_size, void* d_ws, size_t ws_size,
                              hipStream_t stream) {
  const float* x_in   = (const float*)d_in[0];
  const int*   ei     = (const int*)d_in[1];
  const float* Wstart = (const float*)d_in[2];
  const float* bstart = (const float*)d_in[3];
  const float* bnsg   = (const float*)d_in[4];
  const float* bnsb   = (const float*)d_in[5];
  const float* cw1    = (const float*)d_in[6];
  const float* cb1    = (const float*)d_in[7];
  const float* cbng   = (const float*)d_in[8];
  const float* cbnb   = (const float*)d_in[9];
  const float* cw2    = (const float*)d_in[10];
  const float* cb2    = (const float*)d_in[11];
  const float* bng    = (const float*)d_in[12];
  const float* bnb    = (const float*)d_in[13];
  const float* bneg   = (const float*)d_in[14];
  const float* bneb   = (const float*)d_in[15];
  const float* lw     = (const float*)d_in[16];
  const float* lb     = (const float*)d_in[17];

  // Workspace layout (floats): A | B | T (each N*H) | embd (G*384) | stats (256)
  float* A   = (float*)d_ws;
  float* B   = A + (size_t)NN * HH;
  float* T   = B + (size_t)NN * HH;
  float* EMB = T + (size_t)NN * HH;
  float* ST  = EMB + (size_t)GG * (LL * HH);

  const int* src = ei;
  const int* dst = ei + EE;

  auto run_bn = [&](float* p, int ld, int nrows, const float* g, const float* b) {
    zero_kernel<<<1, 256, 0, stream>>>(ST, 2 * HH);
    bn_stats_kernel<<<nrows / 64, 128, 0, stream>>>(p, ld, ST);
    bn_apply_relu_kernel<<<(nrows * HH) / 256, 256, 0, stream>>>(
        p, ld, ST, g, b, 1.0f / (float)nrows);
  };

  // ---- start stage: A = relu(BN(x @ Wstart + bstart)) ----
  gemm_bias_kernel<<<NN / 16, 256, 0, stream>>>(x_in, Wstart, bstart, A);
  run_bn(A, HH, NN, bnsg, bnsb);

  int n = NN;
  for (int i = 0; i < LL; ++i) {
    // h = x + sum_{j->i} x_j   (x in A, h in B)
    copy4_kernel<<<(n * 32) / 256, 256, 0, stream>>>((float4*)B, (const float4*)A, n * 32);
    gin_scatter_kernel<<<EE / 8, 256, 0, stream>>>(A, B, src, dst, i);

    // T = relu(BN(h @ w1 + b1))
    gemm_bias_kernel<<<n / 16, 256, 0, stream>>>(B, cw1 + (size_t)i * HH * HH,
                                                 cb1 + i * HH, T);
    run_bn(T, HH, n, cbng + i * HH, cbnb + i * HH);

    // B = relu(BN(T @ w2 + b2))  -> new pre-pool x
    gemm_bias_kernel<<<n / 16, 256, 0, stream>>>(T, cw2 + (size_t)i * HH * HH,
                                                 cb2 + i * HH, B);
    run_bn(B, HH, n, bng + i * HH, bnb + i * HH);

    // Haar low-pass pooling: A = (B[2k]+B[2k+1])/sqrt(2), n /= 2
    n >>= 1;
    haar_pool_kernel<<<(n * 32) / 256, 256, 0, stream>>>((const float4*)B, (float4*)A,
                                                         n * 32);

    // per-graph sum pool + BN + relu into concat slot i
    const int npg = 128 >> (i + 1);
    graph_pool_kernel<<<(GG * HH) / 256, 256, 0, stream>>>(A, EMB, npg, i * HH);
    run_bn(EMB + i * HH, LL * HH, GG, bneg + i * HH, bneb + i * HH);
  }

  final_linear_kernel<<<(GG * CC + 255) / 256, 256, 0, stream>>>(EMB, lw, lb,
                                                                 (float*)d_out);
}